// GINENet_50096498540960
// MI455X (gfx1250) — compile-verified
//
#include <hip/hip_runtime.h>

typedef __attribute__((ext_vector_type(2))) float v2f;
typedef __attribute__((ext_vector_type(8))) float v8f;

#define N_NODES 100000
#define N_EDGES 1200000
#define DD 64
#define N_GRAPHS 128
#define ROW_BLOCKS (N_NODES / 16)   // 6250, exact
#define LDS_STRIDE 66               // pad 64 -> 66 floats to avoid bank conflicts

// ---------------- copy: h = x (float4) ----------------
__global__ void gine_copy_kernel(const float* __restrict__ src,
                                 float* __restrict__ dst, int n4) {
    int i = blockIdx.x * blockDim.x + threadIdx.x;
    if (i < n4) {
        ((float4*)dst)[i] = ((const float4*)src)[i];
    }
}

// ------------- edge messages: h[dst] += relu(x[src] + e) -------------
// 16 threads per edge, float4 per thread, f32 global atomics.
__global__ void gine_edge_kernel(const float* __restrict__ x,
                                 const int* __restrict__ ei,
                                 const float* __restrict__ ea,
                                 float* __restrict__ h, int nEdges) {
    int gid = blockIdx.x * blockDim.x + threadIdx.x;
    int e = gid >> 4;
    if (e >= nEdges) return;
    int c = (gid & 15) << 2;              // 0..60 step 4
    int src = ei[e];
    int dst = ei[nEdges + e];
    float4 xv = *(const float4*)(x + src * DD + c);
    float4 ev = *(const float4*)(ea + (size_t)e * DD + c);
    float m0 = xv.x + ev.x; m0 = m0 > 0.f ? m0 : 0.f;
    float m1 = xv.y + ev.y; m1 = m1 > 0.f ? m1 : 0.f;
    float m2 = xv.z + ev.z; m2 = m2 > 0.f ? m2 : 0.f;
    float m3 = xv.w + ev.w; m3 = m3 > 0.f ? m3 : 0.f;
    float* hp = h + dst * DD + c;
    atomicAdd(hp + 0, m0);
    atomicAdd(hp + 1, m1);
    atomicAdd(hp + 2, m2);
    atomicAdd(hp + 3, m3);
}

// ------------- node MLP via V_WMMA_F32_16X16X4_F32 -------------
// Xout = relu( relu(H @ W1 + b1) @ W2 + b2 ), H:[N,64], W:[64,64]
// One wave per 16-row tile; 8 waves (256 threads) per block.
__global__ __launch_bounds__(256)
void gine_mlp_kernel(const float* __restrict__ H,
                     const float* __restrict__ W1, const float* __restrict__ b1,
                     const float* __restrict__ W2, const float* __restrict__ b2,
                     float* __restrict__ Xout, int nRowBlocks) {
    __shared__ float lds[8 * 16 * LDS_STRIDE];
    const int wave = threadIdx.x >> 5;
    const int lane = threadIdx.x & 31;
    const int rb = blockIdx.x * 8 + wave;
    if (rb >= nRowBlocks) return;           // wave-uniform: EXEC stays all-1s
    const int lo = lane & 15;               // col-in-tile / row-in-tile
    const int hi = lane >> 4;               // half-wave select
    const int row0 = rb * 16;
    float* myLds = lds + wave * 16 * LDS_STRIDE;

    // Load A block: a[t] covers K = 4t..4t+3.
    // Lane layout (ISA 16x4 f32 A): lanes 0-15 hold K={4t,4t+1}, lanes 16-31 K={4t+2,4t+3}.
    v2f a[16];
    const float* hrow = H + (size_t)(row0 + lo) * DD;
#pragma unroll
    for (int t = 0; t < 16; ++t)
        a[t] = *(const v2f*)(hrow + 4 * t + 2 * hi);

    // GEMM1 + bias + relu -> LDS (converted back to row-major)
#pragma unroll
    for (int nt = 0; nt < 4; ++nt) {
        v8f c = {};
#pragma unroll
        for (int t = 0; t < 16; ++t) {
            const float* wp = W1 + (4 * t + 2 * hi) * DD + nt * 16 + lo;
            v2f b; b[0] = wp[0]; b[1] = wp[DD];
            c = __builtin_amdgcn_wmma_f32_16x16x4_f32(
                    false, a[t], false, b, (short)0, c, false, false);
        }
        float bias = b1[nt * 16 + lo];
#pragma unroll
        for (int r = 0; r < 8; ++r) {
            float v = c[r] + bias;
            v = v > 0.f ? v : 0.f;
            // C/D layout: row = r + 8*hi, col = nt*16 + lo
            myLds[(r + 8 * hi) * LDS_STRIDE + nt * 16 + lo] = v;
        }
    }

    // Reload intermediate as A fragments (same-wave DS ops are in-order).
#pragma unroll
    for (int t = 0; t < 16; ++t)
        a[t] = *(const v2f*)(myLds + lo * LDS_STRIDE + 4 * t + 2 * hi);

    // GEMM2 + bias + relu -> global
#pragma unroll
    for (int nt = 0; nt < 4; ++nt) {
        v8f c = {};
#pragma unroll
        for (int t = 0; t < 16; ++t) {
            const float* wp = W2 + (4 * t + 2 * hi) * DD + nt * 16 + lo;
            v2f b; b[0] = wp[0]; b[1] = wp[DD];
            c = __builtin_amdgcn_wmma_f32_16x16x4_f32(
                    false, a[t], false, b, (short)0, c, false, false);
        }
        float bias = b2[nt * 16 + lo];
#pragma unroll
        for (int r = 0; r < 8; ++r) {
            float v = c[r] + bias;
            v = v > 0.f ? v : 0.f;
            Xout[(size_t)(row0 + r + 8 * hi) * DD + nt * 16 + lo] = v;
        }
    }
}

// ---------------- zero small buffer ----------------
__global__ void gine_zero_kernel(float* __restrict__ p, int n) {
    int i = blockIdx.x * blockDim.x + threadIdx.x;
    if (i < n) p[i] = 0.f;
}

// ------------- pooling: pooled[batch[n]] += x[n] -------------
__global__ void gine_pool_kernel(const float* __restrict__ x,
                                 const int* __restrict__ batch,
                                 float* __restrict__ pooled, int nNodes) {
    int gid = blockIdx.x * blockDim.x + threadIdx.x;
    int n = gid >> 4;
    if (n >= nNodes) return;
    int c = (gid & 15) << 2;
    int g = batch[n];
    float4 xv = *(const float4*)(x + (size_t)n * DD + c);
    float* pp = pooled + g * DD + c;
    atomicAdd(pp + 0, xv.x);
    atomicAdd(pp + 1, xv.y);
    atomicAdd(pp + 2, xv.z);
    atomicAdd(pp + 3, xv.w);
}

// ------------- head: out[g] = pooled[g] . Wout + bout -------------
__global__ void gine_out_kernel(const float* __restrict__ pooled,
                                const float* __restrict__ Wout,
                                const float* __restrict__ bout,
                                float* __restrict__ out, int nGraphs) {
    int g = blockIdx.x * blockDim.x + threadIdx.x;
    if (g >= nGraphs) return;
    float s = 0.f;
#pragma unroll
    for (int k = 0; k < DD; ++k)
        s += pooled[g * DD + k] * Wout[k];
    out[g] = s + bout[0];
}

extern "C" void kernel_launch(void* const* d_in, const int* in_sizes, int n_in,
                              void* d_out, int out_size, void* d_ws, size_t ws_size,
                              hipStream_t stream) {
    const float* x_in = (const float*)d_in[0];
    const int*   ei   = (const int*)d_in[1];
    const float* ea   = (const float*)d_in[2];
    const int*   batch= (const int*)d_in[3];
    const float* W1   = (const float*)d_in[4];
    const float* b1   = (const float*)d_in[5];
    const float* W2   = (const float*)d_in[6];
    const float* b2   = (const float*)d_in[7];
    const float* Wout = (const float*)d_in[8];
    const float* bout = (const float*)d_in[9];
    float* out = (float*)d_out;

    // Workspace: A (x ping), B (h), pooled
    float* A = (float*)d_ws;                       // N_NODES*64
    float* B = A + (size_t)N_NODES * DD;           // N_NODES*64
    float* pooled = B + (size_t)N_NODES * DD;      // 128*64

    const int n4 = N_NODES * DD / 4;               // 1.6M float4s
    const int copyBlocks = (n4 + 255) / 256;
    const int edgeThreads = N_EDGES * 16;
    const int edgeBlocks = (edgeThreads + 255) / 256;
    const int mlpBlocks = (ROW_BLOCKS + 7) / 8;
    const int poolThreads = N_NODES * 16;
    const int poolBlocks = (poolThreads + 255) / 256;

    for (int l = 0; l < 3; ++l) {
        const float* xin = (l == 0) ? x_in : A;
        // h = x  (so h = x + aggr after edge kernel; eps = 0)
        gine_copy_kernel<<<copyBlocks, 256, 0, stream>>>(xin, B, n4);
        gine_edge_kernel<<<edgeBlocks, 256, 0, stream>>>(xin, ei, ea, B, N_EDGES);
        gine_mlp_kernel<<<mlpBlocks, 256, 0, stream>>>(
            B, W1 + (size_t)l * DD * DD, b1 + l * DD,
            W2 + (size_t)l * DD * DD, b2 + l * DD, A, ROW_BLOCKS);
    }

    gine_zero_kernel<<<(N_GRAPHS * DD + 255) / 256, 256, 0, stream>>>(pooled, N_GRAPHS * DD);
    gine_pool_kernel<<<poolBlocks, 256, 0, stream>>>(A, batch, pooled, N_NODES);
    gine_out_kernel<<<1, N_GRAPHS, 0, stream>>>(pooled, Wout, bout, out, N_GRAPHS);
}